// CBOW_11441792876954
// MI455X (gfx1250) — compile-verified
//
#include <hip/hip_runtime.h>
#include <hip/hip_bf16.h>
#include <stdint.h>

// Problem constants (fixed by the reference)
#define DIM   300
#define SEGS  75            // 300 f32 = 75 float4 segments per row (1200 B, exactly 75*16)
#define CTX   10            // 2*WINDOW
#define NEGK  5
#define ROWS  16            // 10 ctx + 1 word + 5 neg
#define WPB   2             // waves per block (wave32)
#define TPB   (WPB * 32)

typedef float v2f __attribute__((ext_vector_type(2)));
typedef float v8f __attribute__((ext_vector_type(8)));

// ---- CDNA5 async global->LDS copy (ASYNCcnt), GVS form: saddr = row base (SGPR pair),
//      per-lane VGPR offset. Inline asm for toolchain portability.
__device__ __forceinline__ void async_cp16_s(uint32_t lds_addr, uint32_t voff, uint64_t sbase) {
    asm volatile("global_load_async_to_lds_b128 %0, %1, %2"
                 :: "v"(lds_addr), "v"(voff), "s"(sbase) : "memory");
}
template <int N>
__device__ __forceinline__ void wait_async() {
    asm volatile("s_wait_asynccnt %0" :: "i"(N) : "memory");
}

__device__ __forceinline__ float dot4(float4 a, float4 b) {
    return a.x * b.x + a.y * b.y + a.z * b.z + a.w * b.w;
}
__device__ __forceinline__ float4 add4(float4 a, float4 b) {
    return make_float4(a.x + b.x, a.y + b.y, a.z + b.z, a.w + b.w);
}

// MySigmoid: sigmoid(x), clipped to 1 above +6 and 0 at/below -6
__device__ __forceinline__ float clipped_sigmoid(float x) {
    float s = 1.0f / (1.0f + __expf(-x));
    float r = (x > 6.0f) ? 1.0f : s;
    return (x > -6.0f) ? r : 0.0f;
}

__global__ __launch_bounds__(TPB) void cbow_loss_kernel(
    const float* __restrict__ emb0, const float* __restrict__ emb1,
    const int* __restrict__ ctx_indices, const int* __restrict__ ctx_lens,
    const int* __restrict__ word_idx, const int* __restrict__ neg_indices,
    const int* __restrict__ neg_mask, float* __restrict__ partials, int nB)
{
    __shared__ float4 lbuf[WPB][ROWS][SEGS];   // 38400 B: per-wave staging of 16 gathered rows
    __shared__ float  wsum[WPB];

    const int lane = threadIdx.x & 31;
    // readfirstlane -> b is SGPR-resident: index loads scalarize, row bases stay on the SALU
    const int wid  = __builtin_amdgcn_readfirstlane((int)(threadIdx.x >> 5));
    const int b    = blockIdx.x * WPB + wid;   // one wave per batch element

    float loss = 0.0f;
    if (b < nB) {
        // 16 source-row base addresses (wave-uniform, scalar)
        const float* rowp[ROWS];
        #pragma unroll
        for (int r = 0; r < CTX; ++r)
            rowp[r] = emb0 + (size_t)ctx_indices[b * CTX + r] * DIM;   // may hit zeroed pad row
        rowp[CTX] = emb1 + (size_t)word_idx[b] * DIM;
        #pragma unroll
        for (int k = 0; k < NEGK; ++k)
            rowp[CTX + 1 + k] = emb1 + (size_t)neg_indices[b * NEGK + k] * DIM;

        // Per-lane byte offsets within a row: segments lane, lane+32, lane+64 (<75)
        const uint32_t off0 = (uint32_t)lane * 16u;
        const uint32_t off1 = off0 + 512u;
        const uint32_t off2 = off0 + 1024u;          // valid only for lane < 11
        const uint32_t lrow0 = (uint32_t)(uintptr_t)&lbuf[wid][0][0];

        // DMA-stage all 16 rows (10 ctx first, then word+neg): 30 + 18 async B128 ops
        #pragma unroll
        for (int r = 0; r < ROWS; ++r) {
            const uint64_t sbase = (uint64_t)(uintptr_t)rowp[r];
            const uint32_t lr    = lrow0 + (uint32_t)r * (SEGS * 16u);
            async_cp16_s(lr + off0, off0, sbase);
            async_cp16_s(lr + off1, off1, sbase);
            if (lane < SEGS - 64) async_cp16_s(lr + off2, off2, sbase);
        }

        // ctx rows done when at most the 18 word/neg ops remain (in-order completion)
        wait_async<18>();

        // Context SUM (scale by 1/len folded into the dots; dot is linear)
        float4 a0 = make_float4(0.f, 0.f, 0.f, 0.f), a1 = a0, a2 = a0;
        #pragma unroll
        for (int r = 0; r < CTX; ++r) {
            a0 = add4(a0, lbuf[wid][r][lane]);
            a1 = add4(a1, lbuf[wid][r][lane + 32]);
            if (lane < SEGS - 64) a2 = add4(a2, lbuf[wid][r][lane + 64]);
        }

        const float inv_len = 1.0f / (float)ctx_lens[b];

        wait_async<0>();   // word + negative rows now resident

        // Six 300-wide inner products: c_sum . {w, n0..n4}; wave32 butterfly reduce
        float ips[6];
        #pragma unroll
        for (int j = 0; j < 6; ++j) {
            const int r = CTX + j;
            float s = dot4(a0, lbuf[wid][r][lane]) + dot4(a1, lbuf[wid][r][lane + 32]);
            if (lane < SEGS - 64) s += dot4(a2, lbuf[wid][r][lane + 64]);
            #pragma unroll
            for (int m = 16; m >= 1; m >>= 1)
                s += __shfl_xor(s, m, 32);
            ips[j] = s * inv_len;
        }

        float pl = clipped_sigmoid(ips[0]);
        float d0 = 1.0f - pl;
        loss = 0.5f * d0 * d0;
        #pragma unroll
        for (int k = 0; k < NEGK; ++k) {
            float nl = clipped_sigmoid(ips[1 + k]) * (float)neg_mask[b * NEGK + k];
            loss += 0.5f * nl * nl;
        }
    }

    if (lane == 0) wsum[wid] = loss;
    __syncthreads();
    if (threadIdx.x == 0) {
        float t = 0.0f;
        #pragma unroll
        for (int w = 0; w < WPB; ++w) t += wsum[w];
        partials[blockIdx.x] = t;     // plain store: no init / no atomics needed
    }
}

// Final reduction using V_WMMA_F32_16X16X4_F32 as a summation engine.
// A = all-ones 16x4, B = 64 input floats, C accumulated across iterations.
// D = 1s x B + C  =>  every D row holds the 16 column-sums of B, so
// sum(all 256 D elements) == 16 * sum(inputs) — independent of operand lane
// layout AND of A/B operand order (both all-ones-weighted), making this exact
// and robust. Divide by 16 (power of two, exact) at the end.
__global__ __launch_bounds__(256) void cbow_reduce_wmma_kernel(
    const float* __restrict__ partials, int n, float* __restrict__ out)
{
    const int lane = threadIdx.x & 31;
    const int wid  = threadIdx.x >> 5;          // 8 waves
    v8f acc = {};
    v2f ones; ones[0] = 1.0f; ones[1] = 1.0f;

    const int stride = 8 * 64;                  // 8 waves x 64 floats per WMMA
    for (int base = wid * 64; base < n; base += stride) {   // wave-uniform trip count
        const int i0 = base + lane;
        const int i1 = base + 32 + lane;
        v2f bv;
        bv[0] = (i0 < n) ? partials[i0] : 0.0f;
        bv[1] = (i1 < n) ? partials[i1] : 0.0f;
        // 8 args: (neg_a, A, neg_b, B, c_mod, C, reuse_a, reuse_b)
        acc = __builtin_amdgcn_wmma_f32_16x16x4_f32(false, ones, false, bv,
                                                    (short)0, acc, false, false);
    }

    float t = acc[0] + acc[1] + acc[2] + acc[3] + acc[4] + acc[5] + acc[6] + acc[7];
    #pragma unroll
    for (int m = 16; m >= 1; m >>= 1) t += __shfl_xor(t, m, 32);
    t *= 0.0625f;                               // undo 16x row replication (exact)

    __shared__ float ws[8];
    if (lane == 0) ws[wid] = t;
    __syncthreads();
    if (threadIdx.x == 0) {
        float r = 0.0f;
        #pragma unroll
        for (int w = 0; w < 8; ++w) r += ws[w];
        out[0] = r;
    }
}

extern "C" void kernel_launch(void* const* d_in, const int* in_sizes, int n_in,
                              void* d_out, int out_size, void* d_ws, size_t ws_size,
                              hipStream_t stream) {
    const float* emb0        = (const float*)d_in[0];
    const float* emb1        = (const float*)d_in[1];
    const int*   ctx_indices = (const int*)d_in[2];
    const int*   ctx_lens    = (const int*)d_in[3];
    const int*   word_idx    = (const int*)d_in[4];
    const int*   neg_indices = (const int*)d_in[5];
    const int*   neg_mask    = (const int*)d_in[6];

    const int nB      = in_sizes[3];                 // ctx_lens has B elements
    const int nblocks = (nB + WPB - 1) / WPB;        // 16384 for B=32768
    float* partials   = (float*)d_ws;                // nblocks floats (64 KB)

    cbow_loss_kernel<<<nblocks, TPB, 0, stream>>>(
        emb0, emb1, ctx_indices, ctx_lens, word_idx, neg_indices, neg_mask,
        partials, nB);
    cbow_reduce_wmma_kernel<<<1, 256, 0, stream>>>(partials, nblocks, (float*)d_out);
}